// Iterative_9174050144279
// MI455X (gfx1250) — compile-verified
//
#include <hip/hip_runtime.h>

#define IMG_H 480
#define IMG_W 640
#define EPSV 1e-9f

// Tile / privatization geometry (H = 8*TILE_H, W = 10*TILE_W exactly)
#define TILE_H 60
#define TILE_W 64
#define HALO   12                 // covers |flow| <= 12  (3 sigma)
#define RH (TILE_H + 2 * HALO + 1) // 85  (floor can go -12, floor+1 can go +13)
#define RW (TILE_W + 2 * HALO + 1) // 89  (odd pitch also spreads LDS banks)
#define RN (RH * RW)               // 7565 cells per plane

// ---------------------------------------------------------------------------
// Kernel 0: zero the global accumulators (out = acc_x/acc_y, ws = acc_w).
// ---------------------------------------------------------------------------
__global__ void zero_accum_kernel(float4* __restrict__ out4, long n_out4,
                                  float4* __restrict__ w4, long n_w4) {
    const float4 z = make_float4(0.f, 0.f, 0.f, 0.f);
    long tid = (long)blockIdx.x * blockDim.x + threadIdx.x;
    long stride = (long)gridDim.x * blockDim.x;
    for (long k = tid; k < n_out4; k += stride) out4[k] = z;
    for (long k = tid; k < n_w4; k += stride) w4[k] = z;
}

// ---------------------------------------------------------------------------
// Kernel 1: tiled bilinear splat with LDS privatization.
//  - workgroup owns a TILE_H x TILE_W tile of source pixels
//  - accumulates (w, w*gy, w*gx) into a haloed LDS region via ds_add_f32
//  - |flow| > HALO outliers fall back to native global_atomic_add_f32
//  - flush: coalesced global atomics, skipping empty cells (w == 0)
// ---------------------------------------------------------------------------
__global__ void __launch_bounds__(256)
splat_tiled_kernel(const float* __restrict__ flow_x,
                   const float* __restrict__ flow_y,
                   const int* __restrict__ p_i,
                   const int* __restrict__ p_tref,
                   float* __restrict__ out,
                   float* __restrict__ accw,
                   int P) {
    __shared__ float lds_w[RN];
    __shared__ float lds_y[RN];
    __shared__ float lds_x[RN];

    const int HW  = IMG_H * IMG_W;
    const int b   = blockIdx.z;
    const int ty0 = blockIdx.y * TILE_H;
    const int tx0 = blockIdx.x * TILE_W;
    const int oy0 = ty0 - HALO;          // LDS region origin (image coords)
    const int ox0 = tx0 - HALO;
    const int tid = threadIdx.x;

    for (int k = tid; k < RN; k += 256) {
        lds_w[k] = 0.f; lds_y[k] = 0.f; lds_x[k] = 0.f;
    }
    __syncthreads();

    const int   i  = *p_i;                         // uniform scalar load
    const float dt = (float)(*p_tref) - (float)i;  // (tref - t0), t0 == i

    const float* fxp = flow_x + ((size_t)b * P + i) * (size_t)HW;
    const float* fyp = flow_y + ((size_t)b * P + i) * (size_t)HW;

    float* oxp = out + (size_t)b * 2 * HW;   // channel 0: x accumulator
    float* oyp = oxp + HW;                   // channel 1: y accumulator
    float* awp = accw + (size_t)b * HW;      // weight accumulator

    for (int p = tid; p < TILE_H * TILE_W; p += 256) {
        int py = p >> 6;          // TILE_W == 64
        int px = p & 63;
        int y = ty0 + py;
        int x = tx0 + px;
        int pix = y * IMG_W + x;

        // CDNA5 speculative prefetch: next loop trip reads 4 rows down.
        __builtin_prefetch(fxp + pix + 4 * IMG_W, 0, 0);
        __builtin_prefetch(fyp + pix + 4 * IMG_W, 0, 0);

        float gx = fxp[pix];
        float gy = fyp[pix];
        float wy = (float)y + dt * gy;
        float wx = (float)x + dt * gx;

        // purge_unfeasible: masked warps contribute nothing anywhere.
        if (!(wy >= 0.f && wy <= (float)(IMG_H - 1) &&
              wx >= 0.f && wx <= (float)(IMG_W - 1)))
            continue;

        float ty = floorf(wy), lxf = floorf(wx);
        float dy = wy - ty,    dx  = wx - lxf;
        int   iy = (int)ty,    ix  = (int)lxf;

        float wgt[4] = { (1.f - dy) * (1.f - dx), (1.f - dy) * dx,
                         dy * (1.f - dx),         dy * dx };
        int   cys[4] = { iy, iy, iy + 1, iy + 1 };
        int   cxs[4] = { ix, ix + 1, ix, ix + 1 };

#pragma unroll
        for (int c = 0; c < 4; ++c) {
            int cy = cys[c], cx = cxs[c];
            // iy,ix >= 0 guaranteed; only the +1 corners can exceed bounds.
            if (cy >= IMG_H || cx >= IMG_W) continue;
            float w = wgt[c];
            int ly = cy - oy0;
            int lx = cx - ox0;
            if (ly >= 0 && ly < RH && lx >= 0 && lx < RW) {
                int l = ly * RW + lx;
                atomicAdd(&lds_w[l], w);          // ds_add_f32
                atomicAdd(&lds_y[l], w * gy);
                atomicAdd(&lds_x[l], w * gx);
            } else {                              // rare far-flier fallback
                int f = cy * IMG_W + cx;
                unsafeAtomicAdd(awp + f, w);      // global_atomic_add_f32
                unsafeAtomicAdd(oyp + f, w * gy);
                unsafeAtomicAdd(oxp + f, w * gx);
            }
        }
    }
    __syncthreads();

    // Flush LDS region to global accumulators (coalesced atomic adds).
    for (int k = tid; k < RN; k += 256) {
        float w = lds_w[k];
        if (w == 0.f) continue;   // w==0 => flow sums are exactly 0 too
        int ly = k / RW;
        int lx = k - ly * RW;
        int cy = oy0 + ly;
        int cx = ox0 + lx;
        if (cy < 0 || cy >= IMG_H || cx < 0 || cx >= IMG_W) continue;
        int f = cy * IMG_W + cx;
        unsafeAtomicAdd(awp + f, w);
        unsafeAtomicAdd(oyp + f, lds_y[k]);
        unsafeAtomicAdd(oxp + f, lds_x[k]);
    }
}

// ---------------------------------------------------------------------------
// Kernel 2: normalize in place: out_c = acc_c / (acc_w + eps). float4 I/O.
// ---------------------------------------------------------------------------
__global__ void normalize_kernel(float4* __restrict__ out4,
                                 const float4* __restrict__ w4,
                                 int B) {
    const int HW4 = (IMG_H * IMG_W) / 4;
    int tid = blockIdx.x * blockDim.x + threadIdx.x;
    if (tid >= B * HW4) return;

    int b = tid / HW4;
    int p = tid - b * HW4;

    float4 wv = w4[(size_t)b * HW4 + p];
    float r0 = 1.f / (wv.x + EPSV);
    float r1 = 1.f / (wv.y + EPSV);
    float r2 = 1.f / (wv.z + EPSV);
    float r3 = 1.f / (wv.w + EPSV);

    size_t oxi = (size_t)b * 2 * HW4 + p;   // channel 0 (x)
    size_t oyi = oxi + HW4;                 // channel 1 (y)

    float4 vx = out4[oxi];
    float4 vy = out4[oyi];
    out4[oxi] = make_float4(vx.x * r0, vx.y * r1, vx.z * r2, vx.w * r3);
    out4[oyi] = make_float4(vy.x * r0, vy.y * r1, vy.z * r2, vy.w * r3);
}

// ---------------------------------------------------------------------------
extern "C" void kernel_launch(void* const* d_in, const int* in_sizes, int n_in,
                              void* d_out, int out_size, void* d_ws, size_t ws_size,
                              hipStream_t stream) {
    const float* flow_x = (const float*)d_in[0];   // (B, P, H, W) f32
    const float* flow_y = (const float*)d_in[1];   // (B, P, H, W) f32
    const int*   p_i    = (const int*)d_in[2];     // scalar
    const int*   p_tref = (const int*)d_in[3];     // scalar

    float* out  = (float*)d_out;                   // (B, 2, H, W) f32
    float* accw = (float*)d_ws;                    // (B, H, W) f32 weight plane

    const int HW = IMG_H * IMG_W;
    const int N  = out_size / 2;                   // B*H*W
    const int B  = N / HW;
    const int P  = in_sizes[0] / N;

    // 1) zero accumulators (kernels accumulate in place; graph replays reuse)
    long n_out4 = (long)out_size / 4;
    long n_w4   = (long)N / 4;
    zero_accum_kernel<<<2048, 256, 0, stream>>>((float4*)out, n_out4,
                                                (float4*)accw, n_w4);

    // 2) tiled bilinear splat (LDS-privatized scatter)
    dim3 grid(IMG_W / TILE_W, IMG_H / TILE_H, B);  // (10, 8, B)
    splat_tiled_kernel<<<grid, 256, 0, stream>>>(
        flow_x, flow_y, p_i, p_tref, out, accw, P);

    // 3) normalize
    int total4 = B * (HW / 4);
    normalize_kernel<<<(total4 + 255) / 256, 256, 0, stream>>>(
        (float4*)out, (const float4*)accw, B);
}